// VectorQuantizerEMA_11845519802890
// MI455X (gfx1250) — compile-verified
//
#include <hip/hip_runtime.h>
#include <hip/hip_bf16.h>

typedef __attribute__((ext_vector_type(2))) float f2;
typedef __attribute__((ext_vector_type(4))) float f4;
typedef __attribute__((ext_vector_type(8))) float v8f;

#define N_TOKENS 131072
#define KEMB     1024
#define DIM      64
#define ROWS_PER_BLOCK 128
#define CHUNK_COLS     128
#define LDS_STRIDE     68   // 64 + 4 pad: keeps 16B/8B alignment, spreads banks (4*col apart)

// d_out flat concat: [0]=loss | [1 .. 1+N*D)=quantized | [1+N*D]=perplexity | [2+N*D ..)=encodings
#define Q_OFF 1
#define P_OFF (1 + N_TOKENS * DIM)
#define E_OFF (2 + N_TOKENS * DIM)

// ---- kernel 1: ||w||^2 per code + zero histogram -------------------------
__global__ void vq_setup(const float* __restrict__ W,
                         float* __restrict__ w2,
                         unsigned* __restrict__ counts) {
    int t = blockIdx.x * blockDim.x + threadIdx.x;   // 0..1023
    const f4* wr = (const f4*)(W + t * DIM);
    float s = 0.f;
#pragma unroll
    for (int i = 0; i < DIM / 4; ++i) {
        f4 v = wr[i];
        s += v[0]*v[0] + v[1]*v[1] + v[2]*v[2] + v[3]*v[3];
    }
    w2[t] = s;
    counts[t] = 0u;
}

// ---- kernel 2: WMMA distances -> argmin -> gather/loss/one-hot -----------
__global__ void __launch_bounds__(256)
vq_main(const float* __restrict__ X,
        const float* __restrict__ W,
        const float* __restrict__ w2g,
        unsigned* __restrict__ counts,
        float* __restrict__ blockLoss,
        float* __restrict__ out)
{
    __shared__ float sW[CHUNK_COLS * LDS_STRIDE];   // ~34.8 KB
    __shared__ float sW2[CHUNK_COLS];
    __shared__ int   sIdx[ROWS_PER_BLOCK];
    __shared__ float sRed[8];

    const int t    = threadIdx.x;
    const int wave = t >> 5;
    const int lane = t & 31;
    const int l16  = lane & 15;
    const int h    = lane >> 4;

    const int rowBase = blockIdx.x * ROWS_PER_BLOCK + wave * 16;

    // A operand: 16x64 X tile for this wave, ISA f32 A layout:
    // lane 16h+m, reg pair kt holds X[m][4kt+2h], X[m][4kt+2h+1]
    const float* aPtr = X + (size_t)(rowBase + l16) * DIM + 2 * h;
    f2 a[16];
#pragma unroll
    for (int kt = 0; kt < 16; ++kt) a[kt] = *(const f2*)(aPtr + 4 * kt);

    float bestV[8];
    int   bestI[8];
#pragma unroll
    for (int r = 0; r < 8; ++r) { bestV[r] = 3.4e38f; bestI[r] = 0; }

    for (int c = 0; c < KEMB / CHUNK_COLS; ++c) {
        __syncthreads();
        // cooperative chunk load: 128 codes x 64 f32 -> LDS (B128 pieces)
        for (int i = t; i < CHUNK_COLS * (DIM / 4); i += 256) {
            int r  = i >> 4;
            int k4 = i & 15;
            f4 v = *(const f4*)(W + (size_t)(c * CHUNK_COLS + r) * DIM + k4 * 4);
            *(f4*)(sW + r * LDS_STRIDE + k4 * 4) = v;
        }
        if (t < CHUNK_COLS) sW2[t] = w2g[c * CHUNK_COLS + t];
        __syncthreads();

        for (int ct = 0; ct < CHUNK_COLS / 16; ++ct) {
            const int colL = ct * 16 + l16;                 // col within chunk
            const float* bPtr = sW + colL * LDS_STRIDE + 2 * h;
            f2 b[16];
#pragma unroll
            for (int kt = 0; kt < 16; ++kt) b[kt] = *(const f2*)(bPtr + 4 * kt);

            v8f acc = {};
#pragma unroll
            for (int kt = 0; kt < 16; ++kt)
                acc = __builtin_amdgcn_wmma_f32_16x16x4_f32(
                        false, a[kt], false, b[kt],
                        (short)0, acc, false, false);

            const float w2c = sW2[ct * 16 + l16];
            const int gcol  = c * CHUNK_COLS + colL;
#pragma unroll
            for (int r = 0; r < 8; ++r) {
                float d = fmaf(-2.f, acc[r], w2c);    // ||x||^2 constant per row: dropped
                if (d < bestV[r]) { bestV[r] = d; bestI[r] = gcol; } // first-min kept
            }
        }
    }

    // argmin across the 16 lanes of each half (C layout: VGPR r = row r / r+8)
#pragma unroll
    for (int r = 0; r < 8; ++r) {
#pragma unroll
        for (int m = 1; m < 16; m <<= 1) {
            float ov = __shfl_xor(bestV[r], m, 32);
            int   oi = __shfl_xor(bestI[r], m, 32);
            if (ov < bestV[r] || (ov == bestV[r] && oi < bestI[r])) {
                bestV[r] = ov; bestI[r] = oi;
            }
        }
    }
    if (l16 == 0) {
        int base = wave * 16 + h * 8;
#pragma unroll
        for (int r = 0; r < 8; ++r) sIdx[base + r] = bestI[r];
    }
    __syncthreads();

    // histogram for perplexity (integer atomics: deterministic)
    if (t < ROWS_PER_BLOCK) atomicAdd(&counts[sIdx[t]], 1u);

    // quantized = W[idx] (the STE output), plus commitment-loss partial
    float lsum = 0.f;
    float* qOut = out + Q_OFF + (size_t)blockIdx.x * ROWS_PER_BLOCK * DIM;
    const float* xBlk = X + (size_t)blockIdx.x * ROWS_PER_BLOCK * DIM;
    for (int j = t; j < ROWS_PER_BLOCK * DIM; j += 256) {
        int row = j >> 6, k = j & 63;
        float wv = W[(size_t)sIdx[row] * DIM + k];
        float xv = xBlk[j];
        __builtin_nontemporal_store(wv, qOut + j);
        float d = wv - xv;
        lsum = fmaf(d, d, lsum);
    }
#pragma unroll
    for (int m = 16; m >= 1; m >>= 1) lsum += __shfl_xor(lsum, m, 32);
    if (lane == 0) sRed[wave] = lsum;
    __syncthreads();
    if (t == 0) {
        float s = 0.f;
#pragma unroll
        for (int wv = 0; wv < 8; ++wv) s += sRed[wv];
        blockLoss[blockIdx.x] = s;          // fixed-order partial: deterministic
    }

    // one-hot encodings: 128 rows x 1024 f32, streaming stores
    float* eBase = out + E_OFF + (size_t)blockIdx.x * ROWS_PER_BLOCK * KEMB;
    for (int row = 0; row < ROWS_PER_BLOCK; ++row) {
        int idx = sIdx[row];
        f2* erow = (f2*)(eBase + (size_t)row * KEMB);
#pragma unroll
        for (int u = 0; u < 2; ++u) {
            int c2 = t + u * 256;                 // 0..511 f2 slots
            f2 v = {0.f, 0.f};
            if ((idx >> 1) == c2) v[idx & 1] = 1.f;
            __builtin_nontemporal_store(v, erow + c2);
        }
    }
}

// ---- kernel 3: reduce loss + perplexity (fixed order: deterministic) -----
__global__ void vq_finalize(const unsigned* __restrict__ counts,
                            const float* __restrict__ blockLoss,
                            float* __restrict__ out)
{
    __shared__ float sE[8], sL[8];
    int t = threadIdx.x, wave = t >> 5, lane = t & 31;
    float e = 0.f, l = 0.f;
#pragma unroll
    for (int u = 0; u < 4; ++u) {
        int k = t + u * 256;
        float p = (float)counts[k] * (1.f / (float)N_TOKENS);
        e += p * logf(p + 1e-10f);
        l += blockLoss[k];
    }
#pragma unroll
    for (int m = 16; m >= 1; m >>= 1) {
        e += __shfl_xor(e, m, 32);
        l += __shfl_xor(l, m, 32);
    }
    if (lane == 0) { sE[wave] = e; sL[wave] = l; }
    __syncthreads();
    if (t == 0) {
        float te = 0.f, tl = 0.f;
        for (int i = 0; i < 8; ++i) { te += sE[i]; tl += sL[i]; }
        out[0]     = 0.25f * (tl / (float)(N_TOKENS * DIM));   // commitment loss
        out[P_OFF] = expf(-te);                                 // perplexity
    }
}

extern "C" void kernel_launch(void* const* d_in, const int* in_sizes, int n_in,
                              void* d_out, int out_size, void* d_ws, size_t ws_size,
                              hipStream_t stream) {
    const float* X = (const float*)d_in[0];   // [131072, 64]
    const float* W = (const float*)d_in[1];   // [1024, 64]
    float* out = (float*)d_out;

    float*    w2        = (float*)d_ws;            // 1024 f32
    float*    blockLoss = w2 + 1024;               // 1024 f32 (one per main block)
    unsigned* counts    = (unsigned*)(w2 + 2048);  // 1024 u32  (12 KB total ws)

    vq_setup   <<<4,    256, 0, stream>>>(W, w2, counts);
    vq_main    <<<1024, 256, 0, stream>>>(X, W, w2, counts, blockLoss, out);
    vq_finalize<<<1,    256, 0, stream>>>(counts, blockLoss, out);
}